// GraphConvLayer_59287728554034
// MI455X (gfx1250) — compile-verified
//
#include <hip/hip_runtime.h>
#include <hip/hip_bf16.h>

#define HIDDEN 128

typedef __attribute__((ext_vector_type(2))) float v2f;
typedef __attribute__((ext_vector_type(8))) float v8f;

// ---------------------------------------------------------------------------
// Kernel 1: edge scatter-add (messages + degree) using native fp32 L2 atomics.
// One wave (32 lanes) per edge; each lane moves 4 consecutive floats (B128
// gather) and issues 4 global_atomic_add_f32. Lane 0 bumps the degree.
// ---------------------------------------------------------------------------
__device__ __forceinline__ void atomic_add_f32(float* p, float v) {
    __hip_atomic_fetch_add(p, v, __ATOMIC_RELAXED, __HIP_MEMORY_SCOPE_AGENT);
}

__global__ void __launch_bounds__(256)
gc_scatter_kernel(const float* __restrict__ x,
                  const int* __restrict__ senders,
                  const int* __restrict__ receivers,
                  float* __restrict__ agg,
                  float* __restrict__ degree,
                  int n_edges) {
    const int lane = threadIdx.x & 31;
    const int edge = (int)((blockIdx.x * blockDim.x + threadIdx.x) >> 5);
    if (edge >= n_edges) return;

    const int s = senders[edge];
    const int r = receivers[edge];

    const float4 v = *(const float4*)(x + (size_t)s * HIDDEN + lane * 4);
    float* dst = agg + (size_t)r * HIDDEN + lane * 4;
    atomic_add_f32(dst + 0, v.x);
    atomic_add_f32(dst + 1, v.y);
    atomic_add_f32(dst + 2, v.z);
    atomic_add_f32(dst + 3, v.w);
    if (lane == 0) atomic_add_f32(degree + r, 1.0f);
}

// ---------------------------------------------------------------------------
// Kernel 2: fused fp32 WMMA GEMM:
//   out = relu( x @ Wself + (agg * 1/max(deg,1)) @ Wmsg + bias )
// One wave owns a 16-row block and all 8 column tiles (16x16 each).
// Weights are staged per-block into LDS in K-pair layout so each B fragment
// is one ds_load_b64:  wp[kp*128 + n] = { W[2kp][n], W[2kp+1][n] }.
// K-loop in steps of 4 using V_WMMA_F32_16X16X4_F32 (exact fp32 path).
// ---------------------------------------------------------------------------
__device__ __forceinline__ void stage_weights_pairs(v2f* __restrict__ wp,
                                                    const float* __restrict__ w,
                                                    int tid) {
    // 64 k-pairs x 128 n = 8192 float2 = 64 KB, 32 iterations per thread.
#pragma unroll 4
    for (int idx = tid; idx < 64 * 128; idx += 256) {
        const int kp = idx >> 7;   // k-pair index: rows 2kp, 2kp+1
        const int n  = idx & 127;
        v2f v;
        v.x = w[(size_t)(2 * kp) * HIDDEN + n];
        v.y = w[(size_t)(2 * kp + 1) * HIDDEN + n];
        wp[idx] = v;
    }
}

__global__ void __launch_bounds__(256)
gc_gemm_kernel(const float* __restrict__ x,
               const float* __restrict__ agg,
               const float* __restrict__ degree,
               const float* __restrict__ wself,
               const float* __restrict__ wmsg,
               const float* __restrict__ bias,
               float* __restrict__ out,
               int n_nodes) {
    __shared__ v2f wp[64 * 128];  // 64 KB: one weight matrix in K-pair layout

    const int tid    = threadIdx.x;
    const int wave   = (int)(blockIdx.x * 8 + (tid >> 5));
    const int lane   = tid & 31;
    const int m_base = wave * 16;
    // Wave-uniform activity flag (n_nodes % 16 == 0, so active waves are full).
    // All waves run staging + barriers; only active waves run WMMA (EXEC all-1s).
    const bool active = (m_base < n_nodes);

    const int half = lane >> 4;   // 0: K pair {k0,k0+1}, 1: K pair {k0+2,k0+3}
    const int l15  = lane & 15;
    const int row  = active ? (m_base + l15) : 0;

    const float* __restrict__ xrow = x   + (size_t)row * HIDDEN;
    const float* __restrict__ arow = agg + (size_t)row * HIDDEN;
    const float invd = 1.0f / fmaxf(degree[row], 1.0f);

    // Accumulators: 8 N-tiles of 16x16 f32 (8 VGPRs each). Init with bias.
    v8f acc[8];
#pragma unroll
    for (int t = 0; t < 8; ++t) {
        const float b = bias[t * 16 + l15];  // C col = lane&15 for every VGPR
#pragma unroll
        for (int i = 0; i < 8; ++i) acc[t][i] = b;
    }

    // ---- stage Wself, pass 1: x @ Wself ----
    stage_weights_pairs(wp, wself, tid);
    __syncthreads();

    if (active) {
#pragma unroll 2
        for (int k0 = 0; k0 < HIDDEN; k0 += 4) {
            const int ka = k0 + half * 2;
            v2f a;
            a.x = xrow[ka];
            a.y = xrow[ka + 1];
            const v2f* wrow = wp + ((k0 >> 1) + half) * 128 + l15;
#pragma unroll
            for (int t = 0; t < 8; ++t) {
                const v2f b = wrow[t * 16];  // ds_load_b64, immediate offset
                acc[t] = __builtin_amdgcn_wmma_f32_16x16x4_f32(
                    false, a, false, b, (short)0, acc[t], false, false);
            }
        }
    }
    __syncthreads();

    // ---- stage Wmsg, pass 2: (agg * 1/deg) @ Wmsg ----
    stage_weights_pairs(wp, wmsg, tid);
    __syncthreads();

    if (active) {
#pragma unroll 2
        for (int k0 = 0; k0 < HIDDEN; k0 += 4) {
            const int ka = k0 + half * 2;
            v2f a;
            a.x = arow[ka] * invd;
            a.y = arow[ka + 1] * invd;
            const v2f* wrow = wp + ((k0 >> 1) + half) * 128 + l15;
#pragma unroll
            for (int t = 0; t < 8; ++t) {
                const v2f b = wrow[t * 16];
                acc[t] = __builtin_amdgcn_wmma_f32_16x16x4_f32(
                    false, a, false, b, (short)0, acc[t], false, false);
            }
        }

        // ---- ReLU + store. C/D: VGPR v, lanes 0-15 -> M=v, lanes 16-31 -> M=v+8
        const int m_off = m_base + half * 8;
#pragma unroll
        for (int t = 0; t < 8; ++t) {
#pragma unroll
            for (int v = 0; v < 8; ++v) {
                const int m = m_off + v;
                const int n = t * 16 + l15;
                out[(size_t)m * HIDDEN + n] = fmaxf(acc[t][v], 0.0f);
            }
        }
    }
}

// ---------------------------------------------------------------------------
extern "C" void kernel_launch(void* const* d_in, const int* in_sizes, int n_in,
                              void* d_out, int out_size, void* d_ws, size_t ws_size,
                              hipStream_t stream) {
    const float* x         = (const float*)d_in[0];
    const int*   senders   = (const int*)d_in[1];
    const int*   receivers = (const int*)d_in[2];
    const float* wself     = (const float*)d_in[3];
    const float* wmsg      = (const float*)d_in[4];
    const float* bias      = (const float*)d_in[5];
    float*       out       = (float*)d_out;

    const int n_nodes = in_sizes[0] / HIDDEN;
    const int n_edges = in_sizes[1];

    // Scratch layout: agg [n_nodes*HIDDEN] fp32, then degree [n_nodes] fp32.
    float* agg    = (float*)d_ws;
    float* degree = agg + (size_t)n_nodes * HIDDEN;
    const size_t zero_bytes = ((size_t)n_nodes * HIDDEN + n_nodes) * sizeof(float);
    hipMemsetAsync(d_ws, 0, zero_bytes, stream);

    // 1 wave per edge, 8 edges per 256-thread block.
    const int scatter_blocks = (n_edges + 7) / 8;
    gc_scatter_kernel<<<scatter_blocks, 256, 0, stream>>>(
        x, senders, receivers, agg, degree, n_edges);

    // 1 wave per 16-row block, 8 waves per 256-thread block.
    const int row_waves   = (n_nodes + 15) / 16;
    const int gemm_blocks = (row_waves + 7) / 8;
    gc_gemm_kernel<<<gemm_blocks, 256, 0, stream>>>(
        x, agg, degree, wself, wmsg, bias, out, n_nodes);
}